// FOF_40389872451729
// MI455X (gfx1250) — compile-verified
//
#include <hip/hip_runtime.h>

// FOF (Fourier Occupancy Field) rasterizer for MI455X / gfx1250.
//
// Strategy: barycentrics and depth are affine in pixel coords, so per-face we
// precompute 4 coefficient rows [const, d/dpx, d/dpy, 0] (sign-of-area and
// 1/area folded in, depth pre-remapped to zt-space). A 16-face x 16-pixel tile
// of each quantity is then one V_WMMA_F32_16X16X4_F32 (A = face coeffs 16x4,
// B = pixel basis [1,px,py,0] 4x16). D layout puts pixel n in lane n (and
// n+16), 8 faces per lane -> per-pixel accumulators stay in 16 VGPRs, one
// shfl_xor(16) reduction at the end. sin(k*pi*zt) k=1..15 via Chebyshev
// recurrence: 1 hw sin + 1 hw cos + 14 FMAs.
//
// Round-2 micro-opts (from disasm): feed v_sin_f32/v_cos_f32 directly in
// hardware "revolution" units (sin(pi*zt) == v_sin(0.5*zt)), killing the
// mul-by-pi + mul-by-1/2pi pair the libm path emits; and fold the 3-way
// inside test into min3(u0,u1,u2) >= 0 (one cmp instead of three + s_ands).

#define HRES   128
#define NFACE  1024
#define NBATCH 2
#define NC     16
#define PI_F   3.14159265358979323846f

typedef __attribute__((ext_vector_type(2))) float v2f;
typedef __attribute__((ext_vector_type(4))) float v4f;
typedef __attribute__((ext_vector_type(8))) float v8f;

// ---------------------------------------------------------------------------
// Kernel 1: per-face affine coefficients.
// coef[face] = 4 x float4: {l0, l1, l2, ztlin} each as [const, px, py, 0],
// with 1/area folded in; ztlin = 0.5*(sum l_i z_i) + 0.5 so the pixel kernel
// only clamps. sgn[face] = sign(area)  (0 for degenerate -> w==0 always).
// ---------------------------------------------------------------------------
__global__ void fof_face_coeffs(const float* __restrict__ v,
                                float* __restrict__ coef,
                                float* __restrict__ sgn) {
  int idx = blockIdx.x * blockDim.x + threadIdx.x;
  if (idx >= NBATCH * NFACE) return;
  const float* vp = v + (size_t)idx * 9;
  // tmp = v * [H/2, -H/2, 1] + [H/2-0.5, H/2-0.5, 0]
  float x0 = fmaf(vp[0],  64.0f, 63.5f);
  float y0 = fmaf(vp[1], -64.0f, 63.5f);
  float z0 = vp[2];
  float x1 = fmaf(vp[3],  64.0f, 63.5f);
  float y1 = fmaf(vp[4], -64.0f, 63.5f);
  float z1 = vp[5];
  float x2 = fmaf(vp[6],  64.0f, 63.5f);
  float y2 = fmaf(vp[7], -64.0f, 63.5f);
  float z2 = vp[8];

  float area = (x1 - x0) * (y2 - y0) - (x2 - x0) * (y1 - y0);
  float inv  = (area != 0.0f) ? (1.0f / area) : 0.0f;
  float s    = (area > 0.0f) ? 1.0f : ((area < 0.0f) ? -1.0f : 0.0f);

  // e0 = (x2-x1)(py-y1) - (y2-y1)(px-x1); l0 = e0/area, etc. (cyclic)
  float b0 = -(y2 - y1) * inv, c0 = (x2 - x1) * inv;
  float a0 = ((y2 - y1) * x1 - (x2 - x1) * y1) * inv;
  float b1 = -(y0 - y2) * inv, c1 = (x0 - x2) * inv;
  float a1 = ((y0 - y2) * x2 - (x0 - x2) * y2) * inv;
  float b2 = -(y1 - y0) * inv, c2 = (x1 - x0) * inv;
  float a2 = ((y1 - y0) * x0 - (x1 - x0) * y0) * inv;

  float az = 0.5f * (a0 * z0 + a1 * z1 + a2 * z2) + 0.5f;
  float bz = 0.5f * (b0 * z0 + b1 * z1 + b2 * z2);
  float cz = 0.5f * (c0 * z0 + c1 * z1 + c2 * z2);

  v4f* cp = (v4f*)(coef + (size_t)idx * 16);
  cp[0] = (v4f){a0, b0, c0, 0.0f};
  cp[1] = (v4f){a1, b1, c1, 0.0f};
  cp[2] = (v4f){a2, b2, c2, 0.0f};
  cp[3] = (v4f){az, bz, cz, 0.0f};
  sgn[idx] = s;
}

// ---------------------------------------------------------------------------
// Kernel 2: WMMA rasterize + Fourier accumulate.
// One wave32 owns 16 pixels (one row segment). Loop over 64 face-tiles of 16.
// A (16x4, f32, 2 VGPR): lane l<16 -> face tile+l, K{0,1}; lane l>=16 -> same
// face, K{2,3}. B (4x16): VGPR0 = rows K0|K2 across lane halves, VGPR1 =
// K1|K3. D (16x16 f32, 8 VGPR): lane l<16 -> rows 0-7 (faces tile+0..7) of
// pixel column l; lane l>=16 -> rows 8-15 of pixel l-16.
// ---------------------------------------------------------------------------
__launch_bounds__(256)
__global__ void fof_pixel(const float* __restrict__ coef,
                          const float* __restrict__ sgn,
                          float* __restrict__ out) {
  const int lane  = threadIdx.x & 31;
  const int wave  = (blockIdx.x << 3) + (threadIdx.x >> 5);  // 0..2047
  const int batch = wave >> 10;
  const int pix16 = wave & 1023;
  const int py    = pix16 >> 3;
  const int pxb   = (pix16 & 7) << 4;
  const int n     = lane & 15;
  const bool hi   = lane >= 16;

  // Pixel-basis B matrix (constant over the whole face loop).
  v2f bmat;
  bmat.x = hi ? (float)py : 1.0f;            // rows K2 | K0
  bmat.y = hi ? 0.0f : (float)(pxb + n);     // rows K3 | K1

  const float* cb = coef + ((size_t)batch * NFACE + n) * 16 + (hi ? 2 : 0);
  const float* sb = sgn + (size_t)batch * NFACE + (hi ? 8 : 0);

  float acc[NC];
#pragma unroll
  for (int k = 0; k < NC; ++k) acc[k] = 0.0f;

  const v8f czero = {};

  for (int tile = 0; tile < NFACE; tile += 16) {
    const float* cf = cb + (size_t)tile * 16;
    v2f a0 = *(const v2f*)(cf + 0);    // l0 coeffs
    v2f a1 = *(const v2f*)(cf + 4);    // l1 coeffs
    v2f a2 = *(const v2f*)(cf + 8);    // l2 coeffs
    v2f a3 = *(const v2f*)(cf + 12);   // ztlin coeffs

    v8f d0 = __builtin_amdgcn_wmma_f32_16x16x4_f32(false, a0, false, bmat,
                                                   (short)0, czero, false, false);
    v8f d1 = __builtin_amdgcn_wmma_f32_16x16x4_f32(false, a1, false, bmat,
                                                   (short)0, czero, false, false);
    v8f d2 = __builtin_amdgcn_wmma_f32_16x16x4_f32(false, a2, false, bmat,
                                                   (short)0, czero, false, false);
    v8f d3 = __builtin_amdgcn_wmma_f32_16x16x4_f32(false, a3, false, bmat,
                                                   (short)0, czero, false, false);

    const v4f* sp = (const v4f*)(sb + tile);   // signs for this lane's 8 faces
    v4f s03 = sp[0];
    v4f s47 = sp[1];

#pragma unroll
    for (int r = 0; r < 8; ++r) {
      float u0 = d0[r], u1 = d1[r], u2 = d2[r];
      float zt = d3[r];
      float sv = (r < 4) ? s03[r] : s47[r - 4];
      // inside <=> min3(u0,u1,u2) >= 0  (one cmp instead of three + s_ands)
      float m = fminf(fminf(u0, u1), u2);
      float w = (m >= 0.0f) ? sv : 0.0f;
      zt = fminf(fmaxf(zt, 0.0f), 1.0f);       // -> single v_max_num ... clamp
      acc[0] = fmaf(w, 1.0f - zt, acc[0]);
      // v_sin_f32/v_cos_f32 take revolutions: sin(pi*zt) == v_sin(0.5*zt).
      float h   = 0.5f * zt;
      float sk  = __builtin_amdgcn_sinf(h);
      float ckv = __builtin_amdgcn_cosf(h);
      float twoc = 2.0f * ckv;
      float skm1 = 0.0f;                       // sin(0)
      acc[1] = fmaf(w, sk, acc[1]);
#pragma unroll
      for (int k = 2; k < NC; ++k) {           // sin(k*theta) recurrence
        float sn = fmaf(twoc, sk, -skm1);
        skm1 = sk;
        sk = sn;
        acc[k] = fmaf(w, sk, acc[k]);
      }
    }
  }

  // Fold the two face-halves: lanes l and l+16 hold the same pixel.
#pragma unroll
  for (int k = 0; k < NC; ++k)
    acc[k] += __shfl_xor(acc[k], 16, 32);

  if (!hi) {
    float* ob = out + (((size_t)batch * NC) * HRES + py) * HRES + pxb + n;
#pragma unroll
    for (int k = 0; k < NC; ++k) {
      float scale = (k == 0) ? 1.0f : (-2.0f / (PI_F * (float)k));
      ob[(size_t)k * HRES * HRES] = acc[k] * scale;
    }
  }
}

// ---------------------------------------------------------------------------
extern "C" void kernel_launch(void* const* d_in, const int* in_sizes, int n_in,
                              void* d_out, int out_size, void* d_ws, size_t ws_size,
                              hipStream_t stream) {
  const float* v = (const float*)d_in[0];   // [B,F,3,3] f32
  // d_in[1] = C (=16), compile-time constant here.
  float* coef = (float*)d_ws;                               // B*F*16 f32 = 128 KiB
  float* sg   = coef + (size_t)NBATCH * NFACE * 16;         // B*F f32   =   8 KiB
  float* out  = (float*)d_out;                              // [B,16,128,128] f32

  fof_face_coeffs<<<(NBATCH * NFACE + 255) / 256, 256, 0, stream>>>(v, coef, sg);

  const int nwaves = NBATCH * (HRES * HRES) / 16;   // 2048 waves
  fof_pixel<<<nwaves / 8, 256, 0, stream>>>(coef, sg, out);
}